// Average_Model_fwRF_54674933678759
// MI455X (gfx1250) — compile-verified
//
#include <hip/hip_runtime.h>

typedef __attribute__((ext_vector_type(2))) float v2f;
typedef __attribute__((ext_vector_type(8))) float v8f;

#define NSEG      8
#define CHUNK     16384        // elements per partial-dot block (64 KB)
#define MAXCHUNKS 9            // ceil(139968/16384)
#define PPB       (NSEG * MAXCHUNKS)   // 72 partials per batch row (multiple of 4)
#define BATCH     512
#define NW        347240       // fused weight vector length

// Segment tables (order: conv0..4, fc0, fc1, fc2)
//  lengths = C*H*H per batch slab; all multiples of 4 -> pure float4 streaming
#define SEG_TABLES \
  constexpr int LEN[8]  = {46656, 139968, 64896, 43264, 43264, 4096, 4096, 1000}; \
  constexpr int BASE[8] = {0, 46656, 186624, 251520, 294784, 338048, 342144, 346240}; \
  constexpr int HH[8]   = {729, 729, 169, 169, 169, 1, 1, 1}; \
  constexpr int WOFF[8] = {0, 64, 256, 640, 896, 1152, 2176, 3200};

// ---------------------------------------------------------------------------
// Kernel 1: build fused weight vector Wm.
//   conv seg i : Wm[base+c*hh+p] = W[woff+c] * mass_i[p]
//   fc0/fc1    : zero-fill (scatter kernel fills selected channels)
//   fc2        : Wm[base+ch] = s * W[3200+ch]
// ---------------------------------------------------------------------------
__global__ __launch_bounds__(256) void build_wm(
    const float* __restrict__ W, const float* __restrict__ mfc,
    const float* __restrict__ m0, const float* __restrict__ m1,
    const float* __restrict__ m2, const float* __restrict__ m3,
    const float* __restrict__ m4, float* __restrict__ Wm) {
  SEG_TABLES
  const int seg = blockIdx.y;
  const int e = blockIdx.x * blockDim.x + threadIdx.x;
  if (e >= LEN[seg]) return;
  float v;
  if (seg < 5) {
    const float* m = (seg == 0) ? m0 : (seg == 1) ? m1 : (seg == 2) ? m2
                   : (seg == 3) ? m3 : m4;
    const int hh = HH[seg];
    const int c = e / hh, p = e - c * hh;
    v = W[WOFF[seg] + c] * m[p];
  } else if (seg == 7) {
    v = mfc[0] * W[WOFF[7] + e];
  } else {
    v = 0.0f;                                  // fc0/fc1 dense slots default 0
  }
  Wm[BASE[seg] + e] = v;
}

// ---------------------------------------------------------------------------
// Kernel 2: scatter selected fc0/fc1 weights (idx sorted & unique -> no races)
// ---------------------------------------------------------------------------
__global__ __launch_bounds__(256) void scatter_fc(
    const float* __restrict__ W, const float* __restrict__ mfc,
    const int* __restrict__ idx0, const int* __restrict__ idx1,
    float* __restrict__ Wm) {
  const int j = blockIdx.x * blockDim.x + threadIdx.x;
  const float s = mfc[0];
  if (j < 1024) {
    Wm[338048 + idx0[j]] = s * W[1152 + j];
  } else if (j < 2048) {
    const int k = j - 1024;
    Wm[342144 + idx1[k]] = s * W[2176 + k];
  }
}

// ---------------------------------------------------------------------------
// Kernel 3: main streaming dot. One block = (batch b, segment seg, chunk).
// Fully coalesced float4 loads of the data slab; Wm chunk is L2-resident.
// Deterministic LDS tree reduction; every (b,seg,chunk) slot is written
// (0 for empty chunks) so partials are fully initialized each call.
// ---------------------------------------------------------------------------
__global__ __launch_bounds__(256) void dot_partial(
    const float* __restrict__ f0, const float* __restrict__ f1,
    const float* __restrict__ f2, const float* __restrict__ f3,
    const float* __restrict__ f4, const float* __restrict__ g0,
    const float* __restrict__ g1, const float* __restrict__ g2,
    const float* __restrict__ Wm, float* __restrict__ partials) {
  SEG_TABLES
  const int chunk = blockIdx.x, seg = blockIdx.y, b = blockIdx.z;
  const int t = threadIdx.x;

  const float* base;
  switch (seg) {
    case 0: base = f0; break;
    case 1: base = f1; break;
    case 2: base = f2; break;
    case 3: base = f3; break;
    case 4: base = f4; break;
    case 5: base = g0; break;
    case 6: base = g1; break;
    default: base = g2; break;
  }
  const int len = LEN[seg];
  const float* __restrict__ X    = base + (size_t)b * (size_t)len;
  const float* __restrict__ Wseg = Wm + BASE[seg];

  float acc = 0.0f;
  const int e0 = chunk * CHUNK;
  if (e0 < len) {
    const int eEnd = min(len, e0 + CHUNK);       // len, CHUNK both %4 == 0
    for (int e = e0 + t * 4; e < eEnd; e += 256 * 4) {
      if (e + 2048 < len) __builtin_prefetch(X + e + 2048, 0, 1);
      const float4 x = *(const float4*)(X + e);
      const float4 w = *(const float4*)(Wseg + e);
      acc = fmaf(x.x, w.x, acc);
      acc = fmaf(x.y, w.y, acc);
      acc = fmaf(x.z, w.z, acc);
      acc = fmaf(x.w, w.w, acc);
    }
  }

  __shared__ float red[256];
  red[t] = acc;
  __syncthreads();
  for (int s2 = 128; s2 > 0; s2 >>= 1) {         // fixed-order tree: deterministic
    if (t < s2) red[t] += red[t + s2];
    __syncthreads();
  }
  if (t == 0)
    partials[((size_t)b * NSEG + seg) * MAXCHUNKS + chunk] = red[0];
}

// ---------------------------------------------------------------------------
// Kernel 4: finish with V_WMMA_F32_16X16X4_F32 (f32-exact).
// A[16x4] = 16 batch rows x 4 partials (documented f32 A layout:
//   lane<16 -> K=0,1 ; lane>=16 -> K=2,3, M = lane&15).
// B = all-ones 4x16 (broadcast, layout-agnostic) -> every D column is the
// row-sum. 18 accumulating WMMA steps over K=72. One wave (32 threads) per
// 16 batches; EXEC all ones as WMMA requires.
// ---------------------------------------------------------------------------
__global__ __launch_bounds__(32) void finish_wmma(
    const float* __restrict__ partials, const float* __restrict__ bias,
    float* __restrict__ out) {
  const int lane = threadIdx.x;                 // 0..31 (wave32)
  const int bb = blockIdx.x * 16;               // batch base
  const int m = lane & 15;
  const int kh = (lane >> 4) * 2;               // 0 for lanes 0-15, 2 for 16-31

  const float* __restrict__ row = partials + (size_t)(bb + m) * PPB;

  v2f bones; bones.x = 1.0f; bones.y = 1.0f;    // B[k,n] = 1 for all k,n
  v8f c = {};
  for (int k = 0; k < PPB; k += 4) {
    v2f a;
    a.x = row[k + kh];
    a.y = row[k + kh + 1];
    // D = A x B + C  (8 args: neg_a, A, neg_b, B, c_mod, C, reuse_a, reuse_b)
    c = __builtin_amdgcn_wmma_f32_16x16x4_f32(
        false, a, false, bones, (short)0, c, false, false);
  }

  const float bv = bias[0];
  // C/D layout: lanes 0-15 hold M = v (N = lane), lanes 16-31 hold M = v+8.
  if (lane == 0) {
    for (int v = 0; v < 8; ++v) out[bb + v] = c[v] + bv;
  } else if (lane == 16) {
    for (int v = 0; v < 8; ++v) out[bb + 8 + v] = c[v] + bv;
  }
}

// ---------------------------------------------------------------------------
// Host-side launch. Inputs in setup_inputs() dict order:
//  0:fmap0 1:mass0 2:fmap1 3:mass1 4:fmap2 5:mass2 6:fmap3 7:mass3
//  8:fmap4 9:mass4 10:fc0 11:fc1 12:fc2 13:mfc 14:idx0 15:idx1 16:W 17:b
// ---------------------------------------------------------------------------
extern "C" void kernel_launch(void* const* d_in, const int* in_sizes, int n_in,
                              void* d_out, int out_size, void* d_ws,
                              size_t ws_size, hipStream_t stream) {
  (void)in_sizes; (void)n_in; (void)out_size; (void)ws_size;

  const float* fmap0 = (const float*)d_in[0];
  const float* mass0 = (const float*)d_in[1];
  const float* fmap1 = (const float*)d_in[2];
  const float* mass1 = (const float*)d_in[3];
  const float* fmap2 = (const float*)d_in[4];
  const float* mass2 = (const float*)d_in[5];
  const float* fmap3 = (const float*)d_in[6];
  const float* mass3 = (const float*)d_in[7];
  const float* fmap4 = (const float*)d_in[8];
  const float* mass4 = (const float*)d_in[9];
  const float* fc0   = (const float*)d_in[10];
  const float* fc1   = (const float*)d_in[11];
  const float* fc2   = (const float*)d_in[12];
  const float* mfc   = (const float*)d_in[13];
  const int*   idx0  = (const int*)d_in[14];
  const int*   idx1  = (const int*)d_in[15];
  const float* W     = (const float*)d_in[16];
  const float* bias  = (const float*)d_in[17];
  float* out = (float*)d_out;

  float* Wm = (float*)d_ws;            // NW floats (1.39 MB, L2-resident)
  float* partials = Wm + NW;           // BATCH * PPB floats (NW*4 is 16B-aligned)

  dim3 g1((139968 + 255) / 256, NSEG, 1);
  build_wm<<<g1, 256, 0, stream>>>(W, mfc, mass0, mass1, mass2, mass3, mass4, Wm);

  scatter_fc<<<8, 256, 0, stream>>>(W, mfc, idx0, idx1, Wm);

  dim3 g2(MAXCHUNKS, NSEG, BATCH);
  dot_partial<<<g2, 256, 0, stream>>>(fmap0, fmap1, fmap2, fmap3, fmap4,
                                      fc0, fc1, fc2, Wm, partials);

  finish_wmma<<<BATCH / 16, 32, 0, stream>>>(partials, bias, out);
}